// LocalMessagePassingNetwork_2559800508852
// MI455X (gfx1250) — compile-verified
//
#include <hip/hip_runtime.h>
#include <hip/hip_bf16.h>

#define NNODES 100000
#define NEDGES 1600000
#define FDIM   64
#define NHEADS 2
#define PROJ   (NHEADS * FDIM)   // 128

typedef float v2f __attribute__((ext_vector_type(2)));
typedef float v8f __attribute__((ext_vector_type(8)));

// Order-preserving float <-> uint key mapping so float segment-max can use
// integer atomicMax. key(x) is monotonic in x; key==0 is an "empty" sentinel
// that decodes to NaN (handled like the reference's isfinite guard).
__device__ __forceinline__ unsigned fkey(float x) {
    unsigned b = __float_as_uint(x);
    return (b & 0x80000000u) ? ~b : (b | 0x80000000u);
}
__device__ __forceinline__ float fdecode(unsigned k) {
    unsigned b = (k & 0x80000000u) ? (k & 0x7fffffffu) : ~k;
    return __uint_as_float(b);
}

// ---------------------------------------------------------------------------
// K1: Xp[N,128] = X[N,64] @ W[64,128] using V_WMMA_F32_16X16X4_F32.
// One wave per 16x16 output tile; K-loop of 16 WMMAs (K=4 each).
// 100000 % 16 == 0 and 128 % 16 == 0 -> every wave is full, EXEC all-ones.
// ---------------------------------------------------------------------------
__global__ void gat_proj_wmma(const float* __restrict__ X,
                              const float* __restrict__ W,
                              float* __restrict__ Xp) {
    const int lane  = threadIdx.x & 31;
    const int wave  = threadIdx.x >> 5;
    const int tile  = blockIdx.x * 8 + wave;      // 6250 blocks * 8 waves = 50000 tiles
    const int mtile = tile >> 3;                  // 6250 M tiles
    const int ntile = tile & 7;                   // 8 N tiles
    const int khalf = lane >> 4;                  // lanes 16-31 hold K+2/K+3
    const int l16   = lane & 15;
    const int row   = mtile * 16 + l16;           // A: M striped over lanes 0-15
    const int col   = ntile * 16 + l16;           // B/C/D: N striped over lanes

    v8f acc = {};
#pragma unroll
    for (int kk = 0; kk < FDIM; kk += 4) {
        const int k0 = kk + 2 * khalf;
        // A 16x4 f32: lane holds {K=k0, K=k0+1} of its row (contiguous -> b64 load)
        v2f a = *(const v2f*)(X + row * FDIM + k0);
        // B 4x16 f32: lane holds {K=k0, K=k0+1} of its column
        v2f b;
        b.x = W[(k0 + 0) * PROJ + col];
        b.y = W[(k0 + 1) * PROJ + col];
        acc = __builtin_amdgcn_wmma_f32_16x16x4_f32(false, a, false, b,
                                                    (short)0, acc, false, false);
    }
    // C/D layout: VGPR v = row M=v (lanes 0-15) / M=8+v (lanes 16-31)
    const int mbase = mtile * 16 + khalf * 8;
#pragma unroll
    for (int v = 0; v < 8; ++v)
        Xp[(mbase + v) * PROJ + col] = acc[v];
}

// ---------------------------------------------------------------------------
// K2: per-node attention terms a_s[n,h] = <Xp[n,h,:], att_src[h,:]> (same a_d).
// One wave per node: coalesced 128-float row read + wave32 shuffle reduction.
// ---------------------------------------------------------------------------
__global__ void gat_node_terms(const float* __restrict__ Xp,
                               const float* __restrict__ att_src,
                               const float* __restrict__ att_dst,
                               float* __restrict__ a_s,
                               float* __restrict__ a_d) {
    const int lane = threadIdx.x & 31;
    const int n    = blockIdx.x * 8 + (threadIdx.x >> 5);
    if (n >= NNODES) return;

    float s0 = 0.f, s1 = 0.f, d0 = 0.f, d1 = 0.f;
#pragma unroll
    for (int j = 0; j < 2; ++j) {
        const int f  = lane + 32 * j;            // feature within a head
        const float x0 = Xp[n * PROJ + f];       // head 0
        const float x1 = Xp[n * PROJ + FDIM + f];// head 1
        s0 += x0 * att_src[f];
        s1 += x1 * att_src[FDIM + f];
        d0 += x0 * att_dst[f];
        d1 += x1 * att_dst[FDIM + f];
    }
#pragma unroll
    for (int off = 16; off > 0; off >>= 1) {
        s0 += __shfl_xor(s0, off, 32);
        s1 += __shfl_xor(s1, off, 32);
        d0 += __shfl_xor(d0, off, 32);
        d1 += __shfl_xor(d1, off, 32);
    }
    if (lane == 0) {
        a_s[n * 2 + 0] = s0;  a_s[n * 2 + 1] = s1;
        a_d[n * 2 + 0] = d0;  a_d[n * 2 + 1] = d1;
    }
}

// ---------------------------------------------------------------------------
// K3: edge logits (LeakyReLU 0.2) + ordered-key atomic segment max over dst.
// ---------------------------------------------------------------------------
__global__ void gat_logits_max(const int* __restrict__ src,
                               const int* __restrict__ dst,
                               const float* __restrict__ a_s,
                               const float* __restrict__ a_d,
                               float* __restrict__ logits,
                               unsigned* __restrict__ segkey) {
    const int e = blockIdx.x * blockDim.x + threadIdx.x;
    if (e >= NEDGES) return;
    const int s = src[e], d = dst[e];
    if ((unsigned)s >= NNODES || (unsigned)d >= NNODES) return;
#pragma unroll
    for (int h = 0; h < NHEADS; ++h) {
        float l = a_s[s * 2 + h] + a_d[d * 2 + h];
        l = (l > 0.f) ? l : 0.2f * l;            // attention LeakyReLU
        logits[e * 2 + h] = l;
        atomicMax(&segkey[d * 2 + h], fkey(l));
    }
}

// ---------------------------------------------------------------------------
// K4: ex = exp(logit - segmax[dst]); alpha buffer <- ex; denom[dst] += ex.
// ---------------------------------------------------------------------------
__global__ void gat_exp_denom(const int* __restrict__ dst,
                              const unsigned* __restrict__ segkey,
                              float* __restrict__ alpha,   // in: logits, out: ex
                              float* __restrict__ denom) {
    const int e = blockIdx.x * blockDim.x + threadIdx.x;
    if (e >= NEDGES) return;
    const int d = dst[e];
    if ((unsigned)d >= NNODES) return;
#pragma unroll
    for (int h = 0; h < NHEADS; ++h) {
        float m = fdecode(segkey[d * 2 + h]);
        if (m != m) m = 0.f;                     // empty-segment sentinel (NaN) -> 0
        const float ex = __expf(alpha[e * 2 + h] - m);
        alpha[e * 2 + h] = ex;
        atomicAdd(&denom[d * 2 + h], ex);
    }
}

// ---------------------------------------------------------------------------
// K5: messages. One wave per edge; head mean folded in (0.5 factor), so each
// lane emits 2 atomic f32 adds covering its two output features. Xp (51 MB)
// and out (25.6 MB) are L2-resident on MI455X (192 MB L2).
// ---------------------------------------------------------------------------
__global__ void gat_scatter(const int* __restrict__ src,
                            const int* __restrict__ dst,
                            const float* __restrict__ Xp,
                            const float* __restrict__ alpha,
                            const float* __restrict__ denom,
                            float* __restrict__ outacc) {
    const int lane = threadIdx.x & 31;
    const int e    = blockIdx.x * 8 + (threadIdx.x >> 5);
    if (e >= NEDGES) return;
    const int s = src[e], d = dst[e];
    if ((unsigned)s >= NNODES || (unsigned)d >= NNODES) return;

    const float sc0 = 0.5f * alpha[e * 2 + 0] / (denom[d * 2 + 0] + 1e-16f);
    const float sc1 = 0.5f * alpha[e * 2 + 1] / (denom[d * 2 + 1] + 1e-16f);

    const int f = lane * 2;                       // features f, f+1
    v2f x0 = *(const v2f*)(Xp + s * PROJ + f);        // head 0
    v2f x1 = *(const v2f*)(Xp + s * PROJ + FDIM + f); // head 1
    atomicAdd(&outacc[d * FDIM + f + 0], sc0 * x0.x + sc1 * x1.x);
    atomicAdd(&outacc[d * FDIM + f + 1], sc0 * x0.y + sc1 * x1.y);
}

// ---------------------------------------------------------------------------
// K6: out = LeakyReLU_{0.01}(out + bias)   (in-place, deterministic)
// ---------------------------------------------------------------------------
__global__ void gat_finalize(float* __restrict__ out,
                             const float* __restrict__ bias) {
    const int i = blockIdx.x * blockDim.x + threadIdx.x;
    if (i >= NNODES * FDIM) return;
    const float x = out[i] + bias[i & (FDIM - 1)];
    out[i] = (x > 0.f) ? x : 0.01f * x;
}

extern "C" void kernel_launch(void* const* d_in, const int* in_sizes, int n_in,
                              void* d_out, int out_size, void* d_ws, size_t ws_size,
                              hipStream_t stream) {
    const float* X       = (const float*)d_in[0];  // [N,64]
    const int*   A       = (const int*)  d_in[1];  // [2,E] (int32 on device)
    const float* W       = (const float*)d_in[2];  // [64,128]
    const float* att_src = (const float*)d_in[3];  // [2,64]
    const float* att_dst = (const float*)d_in[4];  // [2,64]
    const float* bias    = (const float*)d_in[5];  // [64]
    float*       out     = (float*)d_out;          // [N,64]

    // Workspace carve-up (~67 MB of f32)
    float*    ws     = (float*)d_ws;
    float*    Xp     = ws;                                   // N*128
    float*    a_s    = Xp  + (size_t)NNODES * PROJ;          // N*2
    float*    a_d    = a_s + (size_t)NNODES * NHEADS;        // N*2
    unsigned* segkey = (unsigned*)(a_d + (size_t)NNODES * NHEADS); // N*2
    float*    denom  = (float*)(segkey + (size_t)NNODES * NHEADS); // N*2
    float*    alpha  = denom + (size_t)NNODES * NHEADS;      // E*2

    const int* src = A;            // row 0
    const int* dst = A + NEDGES;   // row 1

    // Deterministic re-init every launch (graph-capture safe).
    hipMemsetAsync(out,    0, sizeof(float)    * (size_t)NNODES * FDIM,   stream);
    hipMemsetAsync(segkey, 0, sizeof(unsigned) * (size_t)NNODES * NHEADS, stream);
    hipMemsetAsync(denom,  0, sizeof(float)    * (size_t)NNODES * NHEADS, stream);

    gat_proj_wmma <<<6250,                       256, 0, stream>>>(X, W, Xp);
    gat_node_terms<<<(NNODES + 7) / 8,           256, 0, stream>>>(Xp, att_src, att_dst, a_s, a_d);
    gat_logits_max<<<(NEDGES + 255) / 256,       256, 0, stream>>>(src, dst, a_s, a_d, alpha, segkey);
    gat_exp_denom <<<(NEDGES + 255) / 256,       256, 0, stream>>>(dst, segkey, alpha, denom);
    gat_scatter   <<<(NEDGES + 7) / 8,           256, 0, stream>>>(src, dst, Xp, alpha, denom, out);
    gat_finalize  <<<(NNODES * FDIM + 255) / 256, 256, 0, stream>>>(out, bias);
}